// Filtering_72773925863700
// MI455X (gfx1250) — compile-verified
//
#include <hip/hip_runtime.h>
#include <hip/hip_bf16.h>

typedef __attribute__((ext_vector_type(2))) float v2f;
typedef __attribute__((ext_vector_type(8))) float v8f;

#define BDIM 8
#define MDIM 256
#define NDIM 256
#define IDIM 32
#define NI   (NDIM * IDIM)        // 8192
#define BSTRIDE (MDIM * NI)       // 2,097,152 floats per batch

// ---------------------------------------------------------------------------
// Kernel 0: build the 256x256 circulant band-limiting projector P in d_ws.
// P[r,c] = (1 + 2*sum_{k=1..15} cos(2*pi*k*(r-c)/256)) / 256
// ---------------------------------------------------------------------------
__global__ void build_P_kernel(float* __restrict__ P) {
    int idx = blockIdx.x * blockDim.x + threadIdx.x;   // 65536 elements
    int r = idx >> 8;
    int c = idx & 255;
    float d = (float)(r - c);
    const float w = 6.2831853071795864769f / 256.0f;
    float s = 1.0f;
    #pragma unroll
    for (int k = 1; k < 16; ++k) {
        s += 2.0f * cosf(w * (float)k * d);
    }
    P[idx] = s * (1.0f / 256.0f);
}

// ---------------------------------------------------------------------------
// Kernel 1 (term 1): out[b, m, ni] = sum_k P[m,k] * x[b, k, ni]
//   X_b viewed as 256 x 8192 row-major. Workgroup (8 waves) computes a
//   64 x 256 tile; waves arranged 4(m) x 2(n-half); 8 fp32 16x16 accs/wave.
//   LDS B tile is stored k-grouped-by-4 so each B fragment is ONE aligned
//   ds_load_b64 into an even VGPR pair (no shuffle moves, no bank conflicts).
// ---------------------------------------------------------------------------
__global__ __launch_bounds__(256)
void term1_kernel(const float* __restrict__ x, const float* __restrict__ P,
                  float* __restrict__ out) {
    __shared__ __align__(16) float Pt[64][20];     // P[m0+r][kc+c], row pad 20
    __shared__ __align__(16) float Xt4[4][1024];   // [(k>>2)][n*4 + (k&3)]

    const int b   = blockIdx.z;
    const int m0  = blockIdx.y * 64;
    const int nc0 = blockIdx.x * 256;

    const int t    = threadIdx.x;
    const int lane = t & 31;
    const int wave = t >> 5;        // 0..7
    const int idL  = lane & 15;
    const int hi   = lane >> 4;     // 0 or 1
    const int wm   = wave >> 1;     // 0..3  (16-row group)
    const int wn   = wave & 1;      // 0..1  (128-col group)

    const float* xb = x + (size_t)b * BSTRIDE;

    v8f acc[8];
    #pragma unroll
    for (int q = 0; q < 8; ++q) acc[q] = (v8f){0.f,0.f,0.f,0.f,0.f,0.f,0.f,0.f};

    for (int kc = 0; kc < 256; kc += 16) {
        // stage P chunk: one float4 per thread (b128 load + b128 LDS store)
        {
            const int row  = t >> 2;
            const int colb = (t & 3) * 4;
            const float4 pv = *(const float4*)&P[(m0 + row) * 256 + kc + colb];
            *(float4*)&Pt[row][colb] = pv;
        }
        // stage X chunk: 4 float4 per thread, scatter into k-grouped layout
        #pragma unroll
        for (int j = 0; j < 4; ++j) {
            const int krow = j * 4 + (t >> 6);      // 0..15
            const int colb = (t & 63) * 4;          // 0..252
            const float4 xv =
                *(const float4*)&xb[(size_t)(kc + krow) * NI + nc0 + colb];
            float* dst = &Xt4[krow >> 2][0];
            const int km = krow & 3;
            dst[(colb + 0) * 4 + km] = xv.x;
            dst[(colb + 1) * 4 + km] = xv.y;
            dst[(colb + 2) * 4 + km] = xv.z;
            dst[(colb + 3) * 4 + km] = xv.w;
        }
        __syncthreads();

        #pragma unroll
        for (int kk = 0; kk < 4; ++kk) {
            const int kb = kk * 4;
            const v2f a = *(const v2f*)&Pt[wm * 16 + idL][kb + 2 * hi];
            const float* xrow = &Xt4[kk][0];
            #pragma unroll
            for (int q = 0; q < 8; ++q) {
                const int col = wn * 128 + q * 16 + idL;
                const v2f bf = *(const v2f*)&xrow[col * 4 + 2 * hi];
                acc[q] = __builtin_amdgcn_wmma_f32_16x16x4_f32(
                    false, a, false, bf, (short)0, acc[q], false, false);
            }
        }
        __syncthreads();
    }

    // store: row = m0 + wm*16 + v + 8*hi, col = nc0 + wn*128 + q*16 + idL
    float* ob = out + (size_t)b * BSTRIDE;
    #pragma unroll
    for (int q = 0; q < 8; ++q) {
        const int col = nc0 + wn * 128 + q * 16 + idL;
        #pragma unroll
        for (int v = 0; v < 8; ++v) {
            const int m = m0 + wm * 16 + v + 8 * hi;
            ob[(size_t)m * NI + col] = acc[q][v];
        }
    }
}

// ---------------------------------------------------------------------------
// Kernel 2 (term 2): out[b, m, n, i] += sum_k P[n,k] * x[b, m, k, i]
//   One workgroup per (b, m). Whole x[b,m] (256x32) lives in LDS in the
//   k-grouped-by-4 layout. Wave w owns n in [32w, 32w+32) x all 32 i.
// ---------------------------------------------------------------------------
__global__ __launch_bounds__(256)
void term2_kernel(const float* __restrict__ x, const float* __restrict__ P,
                  float* __restrict__ out) {
    __shared__ __align__(16) float Xm4[64][128];   // [(k>>2)][i*4 + (k&3)]
    __shared__ __align__(16) float Pt2[256][20];   // P[n][kc+c], row pad 20

    const int m = blockIdx.x;
    const int b = blockIdx.y;
    const int t    = threadIdx.x;
    const int lane = t & 31;
    const int wave = t >> 5;
    const int idL  = lane & 15;
    const int hi   = lane >> 4;

    const float* xm = x + (size_t)b * BSTRIDE + (size_t)m * NI;

    // load full 256x32 slab via float4, scatter into k-grouped layout
    #pragma unroll
    for (int j = 0; j < 8; ++j) {
        const int idx4 = j * 256 + t;               // float4 index, 0..2047
        const float4 xv = *(const float4*)&xm[idx4 * 4];
        const int k  = idx4 >> 3;                   // 0..255
        const int ib = (idx4 & 7) * 4;              // 0..28
        float* dst = &Xm4[k >> 2][0];
        const int km = k & 3;
        dst[(ib + 0) * 4 + km] = xv.x;
        dst[(ib + 1) * 4 + km] = xv.y;
        dst[(ib + 2) * 4 + km] = xv.z;
        dst[(ib + 3) * 4 + km] = xv.w;
    }

    v8f acc[4];   // [nt(2)][it(2)]
    #pragma unroll
    for (int q = 0; q < 4; ++q) acc[q] = (v8f){0.f,0.f,0.f,0.f,0.f,0.f,0.f,0.f};

    for (int kc = 0; kc < 256; kc += 16) {
        __syncthreads();   // protect Pt2 readers of previous chunk (and Xm4 fill)
        // stage P chunk: 256x16 floats, 4 float4 per thread
        #pragma unroll
        for (int j = 0; j < 4; ++j) {
            const int idx4 = j * 256 + t;
            const int row  = idx4 >> 2;
            const int colb = (idx4 & 3) * 4;
            *(float4*)&Pt2[row][colb] =
                *(const float4*)&P[row * 256 + kc + colb];
        }
        __syncthreads();

        #pragma unroll
        for (int kk = 0; kk < 4; ++kk) {
            const int kb = kk * 4;
            const float* xrow = &Xm4[(kc >> 2) + kk][0];
            #pragma unroll
            for (int nt = 0; nt < 2; ++nt) {
                const int nrow = wave * 32 + nt * 16 + idL;
                const v2f a = *(const v2f*)&Pt2[nrow][kb + 2 * hi];
                #pragma unroll
                for (int it = 0; it < 2; ++it) {
                    const v2f bf =
                        *(const v2f*)&xrow[(it * 16 + idL) * 4 + 2 * hi];
                    acc[nt * 2 + it] = __builtin_amdgcn_wmma_f32_16x16x4_f32(
                        false, a, false, bf, (short)0, acc[nt * 2 + it],
                        false, false);
                }
            }
        }
    }

    // accumulate into out (kernel 1 already wrote term 1; same stream => ordered)
    float* ob = out + (size_t)b * BSTRIDE + (size_t)m * NI;
    #pragma unroll
    for (int nt = 0; nt < 2; ++nt) {
        #pragma unroll
        for (int it = 0; it < 2; ++it) {
            const int i = it * 16 + idL;
            #pragma unroll
            for (int v = 0; v < 8; ++v) {
                const int n = wave * 32 + nt * 16 + v + 8 * hi;
                ob[n * IDIM + i] += acc[nt * 2 + it][v];
            }
        }
    }
}

// ---------------------------------------------------------------------------
extern "C" void kernel_launch(void* const* d_in, const int* in_sizes, int n_in,
                              void* d_out, int out_size, void* d_ws, size_t ws_size,
                              hipStream_t stream) {
    (void)in_sizes; (void)n_in; (void)out_size; (void)ws_size;
    const float* x = (const float*)d_in[0];
    float* out = (float*)d_out;
    float* P = (float*)d_ws;                 // 256*256*4 = 256 KB of scratch

    build_P_kernel<<<256, 256, 0, stream>>>(P);

    dim3 g1(32 /*n-blocks of 256*/, 4 /*m-blocks of 64*/, BDIM);
    term1_kernel<<<g1, 256, 0, stream>>>(x, P, out);

    dim3 g2(MDIM, BDIM);
    term2_kernel<<<g2, 256, 0, stream>>>(x, P, out);
}